// SetCriterion_44074954391603
// MI455X (gfx1250) — compile-verified
//
#include <hip/hip_runtime.h>

// Problem constants (fixed by setup_inputs)
#define N_BATCH  256
#define S_COARSE 37632
#define F_FINE   3136
#define K_SEL    9408
#define P_TOT    12544           // K_SEL + F_FINE
#define H_SRC    256
#define W_SRC    256
#define H_TGT    512
#define W_TGT    512
#define NBINS    32768           // top 16 bits of |logit| float pattern
#define CHUNK    256
#define CHK_C    147             // 147*256 == 37632 exactly
#define CHK_F    13              // ceil(3136/256)
#define TOT_CH   160             // CHK_C + CHK_F

typedef float v2f __attribute__((ext_vector_type(2)));
typedef float v8f __attribute__((ext_vector_type(8)));

// ---------------------------------------------------------------- sampling --
__device__ __forceinline__ float fetch_zero(const float* __restrict__ img,
                                            int H, int W, int x, int y) {
  if (x < 0 || x >= W || y < 0 || y >= H) return 0.0f;
  return img[(size_t)y * W + x];
}

// Matches F.grid_sample(align_corners=False, padding_mode='zeros') as coded in
// the reference point_sample().
__device__ __forceinline__ float bilinear(const float* __restrict__ img,
                                          int H, int W, float cx, float cy) {
  float x = cx * (float)W - 0.5f;
  float y = cy * (float)H - 0.5f;
  float x0f = floorf(x), y0f = floorf(y);
  int   x0 = (int)x0f,   y0 = (int)y0f;
  float wx = x - x0f,    wy = y - y0f;
  float v00 = fetch_zero(img, H, W, x0,     y0);
  float v01 = fetch_zero(img, H, W, x0 + 1, y0);
  float v10 = fetch_zero(img, H, W, x0,     y0 + 1);
  float v11 = fetch_zero(img, H, W, x0 + 1, y0 + 1);
  return v00 * (1.f - wx) * (1.f - wy) + v01 * wx * (1.f - wy)
       + v10 * (1.f - wx) * wy        + v11 * wx * wy;
}

__device__ __forceinline__ float bce_with_logits(float z, float t) {
  return fmaxf(z, 0.0f) - z * t + log1pf(expf(-fabsf(z)));
}

// ----------------------------------------------------- WMMA wave reduction --
// 32-lane sum through the matrix pipe: A(16x4 f32) holds lane values in K0
// (lanes 0-15 -> rows, K=0; lanes 16-31 -> rows, K=2), B = ones, so
// D[m][n] = v[m] + v[m+16].  Sum the 8 D VGPRs, then fold the two half-waves.
// f32 FMA throughout -> exact-order-deterministic.
__device__ __forceinline__ float wave_sum32(float v) {
  v2f a; a[0] = v;    a[1] = 0.0f;
  v2f b; b[0] = 1.0f; b[1] = 1.0f;
  v8f c = {};
  v8f d = __builtin_amdgcn_wmma_f32_16x16x4_f32(false, a, false, b,
                                                (short)0, c, false, false);
  float s = d[0] + d[1] + d[2] + d[3] + d[4] + d[5] + d[6] + d[7];
  s += __shfl_xor(s, 16, 32);
  return s;
}

// Deterministic block reduction of 4 accumulators (blockDim.x multiple of 32).
__device__ __forceinline__ void block_write_partials4(float b, float p, float l,
                                                      float pl,
                                                      float* __restrict__ dst,
                                                      float* red /*>=32 LDS*/) {
  const int lane = threadIdx.x & 31;
  const int wid  = threadIdx.x >> 5;
  float s0 = wave_sum32(b);
  float s1 = wave_sum32(p);
  float s2 = wave_sum32(l);
  float s3 = wave_sum32(pl);
  if (lane == 0) {
    red[wid * 4 + 0] = s0; red[wid * 4 + 1] = s1;
    red[wid * 4 + 2] = s2; red[wid * 4 + 3] = s3;
  }
  __syncthreads();
  if (threadIdx.x == 0) {
    float t0 = 0.f, t1 = 0.f, t2 = 0.f, t3 = 0.f;
    const int nw = blockDim.x >> 5;
    for (int w = 0; w < nw; ++w) {
      t0 += red[w * 4 + 0]; t1 += red[w * 4 + 1];
      t2 += red[w * 4 + 2]; t3 += red[w * 4 + 3];
    }
    dst[0] = t0; dst[1] = t1; dst[2] = t2; dst[3] = t3;
  }
}

// -------------------------------------------------------------- kernel K1 ---
// One 512-thread block per batch: bilinear-sample all coarse points, store
// logits, build a 32768-bin LDS histogram of the |logit| float bit pattern
// (monotone for non-negative floats), then find the K-th-smallest threshold
// bin and how many elements to take from it.
__global__ __launch_bounds__(512) void k1_coarse_hist(
    const float* __restrict__ src, const float* __restrict__ coords,
    float* __restrict__ logits, int* __restrict__ thrBin,
    int* __restrict__ thrNeed) {
  extern __shared__ unsigned int smem_u[];
  unsigned int* hist = smem_u;          // NBINS
  unsigned int* csum = smem_u + NBINS;  // blockDim.x (512)
  const int n   = blockIdx.x;
  const int tid = threadIdx.x;
  const int nt  = blockDim.x;

  for (int i = tid; i < NBINS; i += nt) hist[i] = 0u;
  __syncthreads();

  const float* img = src    + (size_t)n * (H_SRC * W_SRC);
  const float* cd  = coords + (size_t)n * S_COARSE * 2;
  float*       lg  = logits + (size_t)n * S_COARSE;
  for (int s = tid; s < S_COARSE; s += nt) {
    float v = bilinear(img, H_SRC, W_SRC, cd[2 * s], cd[2 * s + 1]);
    lg[s] = v;
    unsigned int bits = __float_as_uint(fabsf(v));
    atomicAdd(&hist[bits >> 16], 1u);
  }
  __syncthreads();

  const int per = NBINS / 512;  // 64 bins per thread
  unsigned int acc = 0u;
  for (int i = 0; i < per; ++i) acc += hist[tid * per + i];
  csum[tid] = acc;
  __syncthreads();

  if (tid == 0) {
    unsigned int run = 0u;
    int tc = 0;
    for (; tc < nt; ++tc) {
      if (run + csum[tc] >= (unsigned)K_SEL) break;
      run += csum[tc];
    }
    int B = tc * per;
    for (;; ++B) {
      if (run + hist[B] >= (unsigned)K_SEL) break;
      run += hist[B];
    }
    thrBin[n]  = B;
    thrNeed[n] = K_SEL - (int)run;  // how many to take from bin B
  }
}

// -------------------------------------------------------------- kernel K2 ---
// Count threshold-bin members per 256-point chunk (for deterministic ranks).
__global__ __launch_bounds__(CHUNK) void k2_count(
    const float* __restrict__ logits, const int* __restrict__ thrBin,
    int* __restrict__ chunkCnt) {
  const int n  = blockIdx.x / CHK_C;
  const int ch = blockIdx.x % CHK_C;
  const int s  = ch * CHUNK + threadIdx.x;
  const unsigned B = (unsigned)thrBin[n];
  float v = logits[(size_t)n * S_COARSE + s];
  int eq = ((__float_as_uint(fabsf(v)) >> 16) == B) ? 1 : 0;
  unsigned long long m = __ballot(eq);
  __shared__ int wc[CHUNK / 32];
  const int lane = threadIdx.x & 31, wid = threadIdx.x >> 5;
  if (lane == 0) wc[wid] = __popcll(m);
  __syncthreads();
  if (threadIdx.x == 0) {
    int t = 0;
    for (int w = 0; w < CHUNK / 32; ++w) t += wc[w];
    chunkCnt[n * CHK_C + ch] = t;
  }
}

// -------------------------------------------------------------- kernel K3 ---
// Exclusive prefix over the 147 chunk counts of each batch (in place).
__global__ void k3_prefix(int* __restrict__ chunkCnt) {
  if (threadIdx.x == 0) {
    int* p  = chunkCnt + blockIdx.x * CHK_C;
    int run = 0;
    for (int c = 0; c < CHK_C; ++c) { int t = p[c]; p[c] = run; run += t; }
  }
}

// -------------------------------------------------------------- kernel K4 ---
// Deterministic selection (bin < B always; bin == B taken in ascending point
// index up to `need`) fused with label sampling + loss partial accumulation.
// The coarse point logit is reused from K1 (identical sampling to the
// reference's second point_sample of src_masks at imp_coords).
__global__ __launch_bounds__(CHUNK) void k4_coarse_loss(
    const float* __restrict__ logits, const float* __restrict__ coords,
    const float* __restrict__ tgt, const int* __restrict__ thrBin,
    const int* __restrict__ thrNeed, const int* __restrict__ chunkPref,
    float* __restrict__ partials) {
  const int n  = blockIdx.x / CHK_C;
  const int ch = blockIdx.x % CHK_C;
  const int s  = ch * CHUNK + threadIdx.x;
  const int lane = threadIdx.x & 31, wid = threadIdx.x >> 5;
  const unsigned B = (unsigned)thrBin[n];
  const int need   = thrNeed[n];

  float logit = logits[(size_t)n * S_COARSE + s];
  unsigned bin = __float_as_uint(fabsf(logit)) >> 16;
  int below = (bin < B) ? 1 : 0;
  int eq    = (bin == B) ? 1 : 0;

  unsigned long long m  = __ballot(eq);
  unsigned long long lt = (1ull << lane) - 1ull;
  int myIntra = __popcll(m & lt);
  __shared__ int wc[CHUNK / 32];
  if (lane == 0) wc[wid] = __popcll(m);
  __syncthreads();
  int wpre = 0;
  for (int w = 0; w < wid; ++w) wpre += wc[w];
  int rank = chunkPref[n * CHK_C + ch] + wpre + myIntra;
  int sel  = below || (eq && rank < need);

  float bce = 0.f, p = 0.f, l = 0.f, pl = 0.f;
  if (sel) {
    const float* img = tgt + (size_t)n * (H_TGT * W_TGT);
    size_t ci = ((size_t)n * S_COARSE + s) * 2;
    float lab = bilinear(img, H_TGT, W_TGT, coords[ci], coords[ci + 1]);
    bce = bce_with_logits(logit, lab);
    p   = 1.f / (1.f + expf(-logit));
    l   = lab;
    pl  = p * lab;
  }
  __shared__ float red[32];
  block_write_partials4(bce, p, l, pl,
                        partials + (size_t)(n * TOT_CH + ch) * 4, red);
}

// -------------------------------------------------------------- kernel K5 ---
// Fine points: always selected; sample both logit and label, accumulate.
__global__ __launch_bounds__(CHUNK) void k5_fine_loss(
    const float* __restrict__ src, const float* __restrict__ tgt,
    const float* __restrict__ coords, float* __restrict__ partials) {
  const int n  = blockIdx.x / CHK_F;
  const int ch = blockIdx.x % CHK_F;
  const int j  = ch * CHUNK + threadIdx.x;

  float bce = 0.f, p = 0.f, l = 0.f, pl = 0.f;
  if (j < F_FINE) {
    size_t ci = ((size_t)n * F_FINE + j) * 2;
    float cx = coords[ci], cy = coords[ci + 1];
    float logit = bilinear(src + (size_t)n * (H_SRC * W_SRC), H_SRC, W_SRC, cx, cy);
    float lab   = bilinear(tgt + (size_t)n * (H_TGT * W_TGT), H_TGT, W_TGT, cx, cy);
    bce = bce_with_logits(logit, lab);
    p   = 1.f / (1.f + expf(-logit));
    l   = lab;
    pl  = p * lab;
  }
  __shared__ float red[32];
  block_write_partials4(bce, p, l, pl,
                        partials + (size_t)(n * TOT_CH + CHK_C + ch) * 4, red);
}

// -------------------------------------------------------------- kernel K6 ---
// One block, thread n == batch n: fixed-order sum of its 160 chunk partials,
// per-batch dice term, then WMMA wave reduce + fixed-order LDS combine.
__global__ __launch_bounds__(N_BATCH) void k6_final(
    const float* __restrict__ partials, const int* __restrict__ nmPtr,
    float* __restrict__ out) {
  const int n = threadIdx.x;
  float bce = 0.f, p = 0.f, l = 0.f, pl = 0.f;
  const float* base = partials + (size_t)n * TOT_CH * 4;
  for (int c = 0; c < TOT_CH; ++c) {
    bce += base[c * 4 + 0]; p  += base[c * 4 + 1];
    l   += base[c * 4 + 2]; pl += base[c * 4 + 3];
  }
  float dice = 1.f - (2.f * pl + 1.f) / (p + l + 1.f);

  float sb = wave_sum32(bce);
  float sd = wave_sum32(dice);
  __shared__ float red[16];
  const int lane = threadIdx.x & 31, wid = threadIdx.x >> 5;
  if (lane == 0) { red[wid * 2] = sb; red[wid * 2 + 1] = sd; }
  __syncthreads();
  if (threadIdx.x == 0) {
    float tb = 0.f, td = 0.f;
    for (int w = 0; w < N_BATCH / 32; ++w) { tb += red[w * 2]; td += red[w * 2 + 1]; }
    float nm = (float)nmPtr[0];
    out[0] = tb / ((float)P_TOT * nm);  // loss_ce = sum_n mean_pts bce / nm
    out[1] = td / nm;                   // loss_dice
  }
}

// ------------------------------------------------------------------ launch --
// Workspace: logits 36.75MB + thr arrays 2KB + chunk prefix 147KB + partials
// 640KB  (~39.3MB total, all fully overwritten each call -> deterministic).
extern "C" void kernel_launch(void* const* d_in, const int* in_sizes, int n_in,
                              void* d_out, int out_size, void* d_ws,
                              size_t ws_size, hipStream_t stream) {
  (void)in_sizes; (void)n_in; (void)out_size; (void)ws_size;
  const float* src = (const float*)d_in[0];  // (256,1,256,256) f32
  const float* tgt = (const float*)d_in[1];  // (256,1,512,512) f32
  const float* rc  = (const float*)d_in[2];  // (256,37632,2)   f32
  const float* rf  = (const float*)d_in[3];  // (256,3136,2)    f32
  const int*   nm  = (const int*)d_in[4];    // scalar int
  float* out = (float*)d_out;                // [loss_ce, loss_dice]

  char* ws = (char*)d_ws;
  float* logits   = (float*)ws; ws += (size_t)N_BATCH * S_COARSE * sizeof(float);
  int*   thrBin   = (int*)ws;   ws += (size_t)N_BATCH * sizeof(int);
  int*   thrNeed  = (int*)ws;   ws += (size_t)N_BATCH * sizeof(int);
  int*   chunkCnt = (int*)ws;   ws += (size_t)N_BATCH * CHK_C * sizeof(int);
  float* partials = (float*)ws; ws += (size_t)N_BATCH * TOT_CH * 4 * sizeof(float);

  const size_t shmem1 = (NBINS + 512) * sizeof(unsigned int);  // 130KB LDS
  k1_coarse_hist<<<N_BATCH, 512, shmem1, stream>>>(src, rc, logits, thrBin, thrNeed);
  k2_count<<<N_BATCH * CHK_C, CHUNK, 0, stream>>>(logits, thrBin, chunkCnt);
  k3_prefix<<<N_BATCH, 32, 0, stream>>>(chunkCnt);
  k4_coarse_loss<<<N_BATCH * CHK_C, CHUNK, 0, stream>>>(logits, rc, tgt, thrBin,
                                                        thrNeed, chunkCnt, partials);
  k5_fine_loss<<<N_BATCH * CHK_F, CHUNK, 0, stream>>>(src, tgt, rf, partials);
  k6_final<<<1, N_BATCH, 0, stream>>>(partials, nm, out);
}